// TimeVaryingBiquad_82935818486359
// MI455X (gfx1250) — compile-verified
//
#include <hip/hip_runtime.h>
#include <stdint.h>

// Problem geometry (fixed by the reference setup).
#define B_DIM 64
#define T_DIM 65536
#define NTOT (B_DIM * T_DIM)       // 4,194,304 samples
#define CHUNK 512                  // samples per wave-chunk
#define NCH (T_DIM / CHUNK)        // 128 chunks per row
#define SEG (CHUNK / 32)           // 16 samples per lane
#define WAVES_PER_BLOCK 4
#define BLOCK_THREADS 128

// d_out layout (floats), concatenated reference outputs:
//   [0,    N) : y_ab
//   [N,   4N) : a_full  [B,T,3] = {1, a1, a2}
//   [4N,  7N) : b_coeff [B,T,3]
//   [7N,  8N) : y_a

// log-space interpolation constants
#define LNW0 (-4.6051702f)   // ln(0.01)
#define LNWR (5.7499001f)    // ln(pi) - ln(0.01)
#define LNQ0 (-0.34658319f)  // ln(0.7071)
#define LNQR (1.7328776f)    // ln(4) - ln(0.7071)
#define SF   (0.999f)        // 1 - EPS

// ---------------------------------------------------------------------------
// CDNA5 async global->LDS DMA (ASYNCcnt-tracked), via inline asm (portable
// across toolchains). LDS byte address = low 32 bits of flat shared pointer.
// Async loads complete IN ORDER, so partial waits release earlier batches.
// ---------------------------------------------------------------------------
__device__ __forceinline__ void async_copy_b32(const float* gsrc, float* ldst) {
  uint32_t laddr = (uint32_t)(uintptr_t)ldst;
  uint64_t gaddr = (uint64_t)(uintptr_t)gsrc;
  asm volatile("global_load_async_to_lds_b32 %0, %1, off"
               :: "v"(laddr), "v"(gaddr)
               : "memory");
}
__device__ __forceinline__ void wait_async0()  { asm volatile("s_wait_asynccnt 0"  ::: "memory"); }
__device__ __forceinline__ void wait_async16() { asm volatile("s_wait_asynccnt 16" ::: "memory"); }
__device__ __forceinline__ void wait_async48() { asm volatile("s_wait_asynccnt 48" ::: "memory"); }

// Affine-map scan: a map is (M, u): s -> M*s + u  with s = (y[t], y[t-1]).
// Per-sample step: M_t = [[-a1,-a2],[1,0]], u_t = (x_t, 0).

// ---------------------------------------------------------------------------
// Pass A (fused coeff + summarize): one wave per (row, chunk).
//  1. async-stage wm, qm (then x) into LDS
//  2. coalesced per-sample coefficient math (v_exp/v_sin/v_cos/v_rcp; no
//     IEEE-div sequences: 1/q folded into the exp, 1/a0 via v_rcp_f32),
//     writing a_full & b_coeff to global and a1/a2 into LDS
//  3. lane-serial affine composition over 16-sample segments + 5-step
//     wave32 shfl_up scan -> chunk summary (H, v) to workspace
// ---------------------------------------------------------------------------
__global__ __launch_bounds__(BLOCK_THREADS)
void coeff_summarize_kernel(const float* __restrict__ x,
                            const float* __restrict__ wm,
                            const float* __restrict__ qm,
                            float* __restrict__ a_full,
                            float* __restrict__ bco,
                            float* __restrict__ ws_sum) {
  // [0:512) wm | [512:1024) qm | [1024:1536) x | [1536:2048) a1 | [2048:2560) a2
  __shared__ float lds[WAVES_PER_BLOCK][5 * CHUNK];  // 40 KB/block
  const int wib  = threadIdx.x >> 5;
  const int lane = threadIdx.x & 31;
  const int wave = blockIdx.x * WAVES_PER_BLOCK + wib;
  const int row  = wave / NCH;
  const int ch   = wave % NCH;
  const size_t g0 = (size_t)row * T_DIM + (size_t)ch * CHUNK;
  float* L = lds[wib];

  // Batch 1 (32 issues): wm, qm. Batch 2 (16 issues): x.
  for (int j = 0; j < CHUNK / 32; ++j) {
    int idx = j * 32 + lane;
    async_copy_b32(&wm[g0 + idx], &L[0 * CHUNK + idx]);
    async_copy_b32(&qm[g0 + idx], &L[1 * CHUNK + idx]);
  }
  for (int j = 0; j < CHUNK / 32; ++j) {
    int idx = j * 32 + lane;
    async_copy_b32(&x[g0 + idx], &L[2 * CHUNK + idx]);
  }
  wait_async16();  // wm/qm landed; x still in flight behind the math below.

  // Coefficient math, coalesced order; global stores are wave-contiguous.
  float* af = a_full + g0 * 3;
  float* bc = bco + g0 * 3;
  for (int j = 0; j < CHUNK / 32; ++j) {
    int idx = j * 32 + lane;
    float wmv = L[0 * CHUNK + idx];
    float qmv = L[1 * CHUNK + idx];
    float w  = __expf(fmaf(LNWR, wmv, LNW0));    // w  in [0.01, pi]
    float iq = __expf(-fmaf(LNQR, qmv, LNQ0));   // 1/q, no division
    float sw, cw;
    __sincosf(w, &sw, &cw);
    float alpha = sw * 0.5f * iq;
    float ia0 = __builtin_amdgcn_rcpf(1.0f + alpha);  // v_rcp_f32
    float a1  = (-2.0f * cw * SF) * ia0;
    float a2  = ((1.0f - alpha) * SF) * ia0;
    float b1v = (1.0f - cw) * ia0;
    float b0v = 0.5f * b1v;
    af[idx * 3 + 0] = 1.0f;
    af[idx * 3 + 1] = a1;
    af[idx * 3 + 2] = a2;
    bc[idx * 3 + 0] = b0v;
    bc[idx * 3 + 1] = b1v;
    bc[idx * 3 + 2] = b0v;          // b2 == b0 for LP
    L[3 * CHUNK + idx] = a1;
    L[4 * CHUNK + idx] = a2;
  }
  wait_async0();                     // x landed
  asm volatile("" ::: "memory");     // order LDS writes before cross-lane reads

  // Lane-serial composition over this lane's 16-sample segment.
  float m00 = 1.f, m01 = 0.f, m10 = 0.f, m11 = 1.f, u0 = 0.f, u1 = 0.f;
  const int s0i = lane * SEG;
  for (int i = 0; i < SEG; ++i) {
    float xv = L[2 * CHUNK + s0i + i];
    float a1 = L[3 * CHUNK + s0i + i];
    float a2 = L[4 * CHUNK + s0i + i];
    float t00 = -a1 * m00 - a2 * m10;
    float t01 = -a1 * m01 - a2 * m11;
    float tu0 = xv - a1 * u0 - a2 * u1;
    m10 = m00; m11 = m01; u1 = u0;
    m00 = t00; m01 = t01; u0 = tu0;
  }
  // Inclusive wave32 scan (ordered composition: mine AFTER earlier lanes').
  for (int d = 1; d < 32; d <<= 1) {
    float o00 = __shfl_up(m00, d, 32), o01 = __shfl_up(m01, d, 32);
    float o10 = __shfl_up(m10, d, 32), o11 = __shfl_up(m11, d, 32);
    float ou0 = __shfl_up(u0,  d, 32), ou1 = __shfl_up(u1,  d, 32);
    if (lane >= d) {
      float c00 = m00 * o00 + m01 * o10;
      float c01 = m00 * o01 + m01 * o11;
      float c10 = m10 * o00 + m11 * o10;
      float c11 = m10 * o01 + m11 * o11;
      float cu0 = m00 * ou0 + m01 * ou1 + u0;
      float cu1 = m10 * ou0 + m11 * ou1 + u1;
      m00 = c00; m01 = c01; m10 = c10; m11 = c11; u0 = cu0; u1 = cu1;
    }
  }
  if (lane == 31) {
    float* s = ws_sum + (size_t)wave * 6;
    s[0] = m00; s[1] = m01; s[2] = m10; s[3] = m11; s[4] = u0; s[5] = u1;
  }
}

// ---------------------------------------------------------------------------
// Pass B: tiny per-row serial scan over the 128 chunk summaries; records the
// carry-in state (y[-1], y[-2]) per chunk. Prefetches ahead (global_prefetch).
// ---------------------------------------------------------------------------
__global__ void chunk_scan_kernel(const float* __restrict__ ws_sum,
                                  float* __restrict__ ws_state) {
  int row = blockIdx.x * blockDim.x + threadIdx.x;
  if (row >= B_DIM) return;
  float s0 = 0.f, s1 = 0.f;
  for (int k = 0; k < NCH; ++k) {
    size_t id = (size_t)row * NCH + k;
    const float* h = ws_sum + id * 6;
    if (k + 8 < NCH) __builtin_prefetch(h + 8 * 6, 0, 1);
    ws_state[id * 2 + 0] = s0;
    ws_state[id * 2 + 1] = s1;
    float n0 = h[0] * s0 + h[1] * s1 + h[4];
    float n1 = h[2] * s0 + h[3] * s1 + h[5];
    s0 = n0; s1 = n1;
  }
}

// ---------------------------------------------------------------------------
// Pass C (apply): one wave per (row, chunk). Async-stage x + the chunk's
// CONTIGUOUS a_full and b triplets (1536 floats each). Exclusive affine scan
// gives every lane its exact (y[t-1], y[t-2]); lanes replay 16 samples
// computing y_a and the FIR y_ab, assembled in LDS and stored coalesced.
// b staging overlaps the scan phase via partial s_wait_asynccnt.
// ---------------------------------------------------------------------------
__global__ __launch_bounds__(BLOCK_THREADS)
void apply_kernel(const float* __restrict__ x,
                  const float* __restrict__ a_full,
                  const float* __restrict__ bco,
                  float* __restrict__ yab,
                  float* __restrict__ ya,
                  const float* __restrict__ ws_state) {
  // [0:512) x | [512:2048) a_full triplets | [2048:3584) b triplets
  __shared__ float lds[WAVES_PER_BLOCK][7 * CHUNK];  // 56 KB/block
  const int wib  = threadIdx.x >> 5;
  const int lane = threadIdx.x & 31;
  const int wave = blockIdx.x * WAVES_PER_BLOCK + wib;
  const int row  = wave / NCH;
  const int ch   = wave % NCH;
  const size_t g0 = (size_t)row * T_DIM + (size_t)ch * CHUNK;
  float* L = lds[wib];

  const float* afc = a_full + g0 * 3;  // 1536 contiguous floats
  const float* bch = bco + g0 * 3;     // 1536 contiguous floats
  // Batch 1 (64 issues): x + a_full. Batch 2 (48 issues): b.
  for (int j = 0; j < CHUNK / 32; ++j) {
    int idx = j * 32 + lane;
    async_copy_b32(&x[g0 + idx], &L[0 * CHUNK + idx]);
  }
  for (int j = 0; j < (3 * CHUNK) / 32; ++j) {
    int idx = j * 32 + lane;
    async_copy_b32(&afc[idx], &L[1 * CHUNK + idx]);
  }
  for (int j = 0; j < (3 * CHUNK) / 32; ++j) {
    int idx = j * 32 + lane;
    async_copy_b32(&bch[idx], &L[4 * CHUNK + idx]);
  }
  wait_async48();  // x + a_full landed; b overlaps the scan below.

  // Lane-serial composition (a1/a2 read from staged triplets, stride 3).
  float m00 = 1.f, m01 = 0.f, m10 = 0.f, m11 = 1.f, u0 = 0.f, u1 = 0.f;
  const int s0i = lane * SEG;
  for (int i = 0; i < SEG; ++i) {
    int idx = s0i + i;
    float xv = L[0 * CHUNK + idx];
    float a1 = L[1 * CHUNK + idx * 3 + 1];
    float a2 = L[1 * CHUNK + idx * 3 + 2];
    float t00 = -a1 * m00 - a2 * m10;
    float t01 = -a1 * m01 - a2 * m11;
    float tu0 = xv - a1 * u0 - a2 * u1;
    m10 = m00; m11 = m01; u1 = u0;
    m00 = t00; m01 = t01; u0 = tu0;
  }
  // Inclusive scan ...
  for (int d = 1; d < 32; d <<= 1) {
    float o00 = __shfl_up(m00, d, 32), o01 = __shfl_up(m01, d, 32);
    float o10 = __shfl_up(m10, d, 32), o11 = __shfl_up(m11, d, 32);
    float ou0 = __shfl_up(u0,  d, 32), ou1 = __shfl_up(u1,  d, 32);
    if (lane >= d) {
      float c00 = m00 * o00 + m01 * o10;
      float c01 = m00 * o01 + m01 * o11;
      float c10 = m10 * o00 + m11 * o10;
      float c11 = m10 * o01 + m11 * o11;
      float cu0 = m00 * ou0 + m01 * ou1 + u0;
      float cu1 = m10 * ou0 + m11 * ou1 + u1;
      m00 = c00; m01 = c01; m10 = c10; m11 = c11; u0 = cu0; u1 = cu1;
    }
  }
  // ... shifted to exclusive (lane 0 = identity).
  float e00 = __shfl_up(m00, 1, 32), e01 = __shfl_up(m01, 1, 32);
  float e10 = __shfl_up(m10, 1, 32), e11 = __shfl_up(m11, 1, 32);
  float eu0 = __shfl_up(u0, 1, 32),  eu1 = __shfl_up(u1, 1, 32);
  if (lane == 0) { e00 = 1.f; e01 = 0.f; e10 = 0.f; e11 = 1.f; eu0 = 0.f; eu1 = 0.f; }

  // Chunk carry-in -> this lane's exact (y[t-1], y[t-2]).
  const float* st = ws_state + ((size_t)row * NCH + ch) * 2;
  float cs0 = st[0], cs1 = st[1];
  float y1 = e00 * cs0 + e01 * cs1 + eu0;
  float y2 = e10 * cs0 + e11 * cs1 + eu1;

  wait_async0();  // b triplets resident.

  // Replay: LPC + FIR. y_a overwrites the consumed x slot; y_ab goes into the
  // never-read "1.0" component slot of the staged a_full triplet.
  for (int i = 0; i < SEG; ++i) {
    int idx = s0i + i;
    float xv = L[0 * CHUNK + idx];
    float a1 = L[1 * CHUNK + idx * 3 + 1];
    float a2 = L[1 * CHUNK + idx * 3 + 2];
    float b0 = L[4 * CHUNK + idx * 3 + 0];
    float b1 = L[4 * CHUNK + idx * 3 + 1];
    float b2 = L[4 * CHUNK + idx * 3 + 2];
    float y  = xv - a1 * y1 - a2 * y2;
    float yo = b0 * y + b1 * y1 + b2 * y2;
    L[0 * CHUNK + idx]         = y;   // y_a
    L[1 * CHUNK + idx * 3 + 0] = yo;  // y_ab (slot held the constant 1.0)
    y2 = y1; y1 = y;
  }
  asm volatile("" ::: "memory");  // LDS stores before cross-lane readback

  // Coalesced writeback; stride-3 LDS reads are conflict-free (gcd(3,64)=1).
  for (int j = 0; j < CHUNK / 32; ++j) {
    int idx = j * 32 + lane;
    ya[g0 + idx]  = L[0 * CHUNK + idx];
    yab[g0 + idx] = L[1 * CHUNK + idx * 3 + 0];
  }
}

extern "C" void kernel_launch(void* const* d_in, const int* in_sizes, int n_in,
                              void* d_out, int out_size, void* d_ws, size_t ws_size,
                              hipStream_t stream) {
  (void)in_sizes; (void)n_in; (void)out_size; (void)ws_size;
  const float* x  = (const float*)d_in[0];
  const float* wm = (const float*)d_in[1];
  const float* qm = (const float*)d_in[2];
  float* out = (float*)d_out;
  const size_t N = (size_t)NTOT;
  float* yab   = out;
  float* afull = out + N;
  float* bco   = out + 4 * N;
  float* ya    = out + 7 * N;

  float* wsf      = (float*)d_ws;                   // 256 KB used total
  float* ws_sum   = wsf;                            // B*NCH*6 floats
  float* ws_state = wsf + (size_t)B_DIM * NCH * 6;  // B*NCH*2 floats

  const int nblocks = (B_DIM * NCH) / WAVES_PER_BLOCK;  // 2048
  coeff_summarize_kernel<<<nblocks, BLOCK_THREADS, 0, stream>>>(
      x, wm, qm, afull, bco, ws_sum);
  chunk_scan_kernel<<<1, 64, 0, stream>>>(ws_sum, ws_state);
  apply_kernel<<<nblocks, BLOCK_THREADS, 0, stream>>>(
      x, afull, bco, yab, ya, ws_state);
}